// CircuitModel_26027501813778
// MI455X (gfx1250) — compile-verified
//
#include <hip/hip_runtime.h>
#include <hip/hip_bf16.h>

// CDNA5 / gfx1250: wave32, WMMA f32 16x16x4 (full-precision matrix op).
//
// Sequential scan (T=2048):  y = sigmoid(W @ x);  W += lr*(th0*y x^T + th1*y^2*W)
// Rows of W are independent recurrences -> parallelize over 512 rows.
// W is register-resident for the whole scan; x_t streamed from L2 (4 MB resident).
// Matvec via V_WMMA_F32_16X16X4_F32 with x replicated across M in the A operand
// and 16 W rows in the B operand => every D row identical => lane L holds
// y[L&15] in c[0] directly (no cross-lane shuffles).
//
// Perf structure (latency-bound problem, ~3.8 GFLOP total):
//  - 32 WGs x 16 rows, 8 waves split K (64 K each -> 16 WMMA/step/wave)
//  - dual WMMA accumulator chains (depth 8 instead of 16)
//  - software prefetch of x_{t+1} into a second register buffer, issued right
//    after the WMMA chain so L2 latency hides behind barrier+reduce+update
//  - double-buffered LDS partials -> exactly one barrier per step

typedef __attribute__((ext_vector_type(2))) float v2f;
typedef __attribute__((ext_vector_type(8))) float v8f;

constexpr int T_STEPS     = 2048;
constexpr int NIN         = 512;
constexpr int NOBS        = 256;
constexpr int ROWS_PER_WG = 16;            // WMMA M/N
constexpr int WAVES       = 8;             // split-K across waves in a WG
constexpr int KPW         = NIN / WAVES;   // 64 K per wave
constexpr int CHUNKS      = KPW / 4;       // 16 WMMA (K=4) per wave per step

__device__ __forceinline__ void scan_step(
    int t, const float* __restrict__ X, float* __restrict__ out,
    float (&part)[2][WAVES * ROWS_PER_WG],
    v2f (&wfrag)[CHUNKS], v2f (&xcur)[CHUNKS], v2f (&xnext)[CHUNKS],
    int wave, int kh, int n, int kbase, int lo, int hi,
    float a_coef, float b_coef)
{
    // ---- matvec partial: two interleaved accumulator chains (depth 8) ----
    v8f acc0 = {};
    v8f acc1 = {};
#pragma unroll
    for (int c = 0; c < CHUNKS; c += 2) {
        acc0 = __builtin_amdgcn_wmma_f32_16x16x4_f32(
                 false, xcur[c],     false, wfrag[c],     (short)0, acc0, false, false);
        acc1 = __builtin_amdgcn_wmma_f32_16x16x4_f32(
                 false, xcur[c + 1], false, wfrag[c + 1], (short)0, acc1, false, false);
    }
    // all D rows identical => lane holds partial dot for column n
    if (kh == 0) part[t & 1][wave * ROWS_PER_WG + n] = acc0[0] + acc1[0];

    // ---- prefetch next step's x slice (independent of scan state) ----
    if (t + 1 < T_STEPS) {
        const float* xp = X + (size_t)(t + 1) * NIN + kbase + 2 * kh;
#pragma unroll
        for (int c = 0; c < CHUNKS; ++c)
            xnext[c] = *(const v2f*)(xp + 4 * c);
    }

    __syncthreads();

    // ---- cross-wave reduction (balanced tree) + sigmoid ----
    const float* p = part[t & 1];
    const float s01 = p[0 * 16 + n] + p[1 * 16 + n];
    const float s23 = p[2 * 16 + n] + p[3 * 16 + n];
    const float s45 = p[4 * 16 + n] + p[5 * 16 + n];
    const float s67 = p[6 * 16 + n] + p[7 * 16 + n];
    const float s   = (s01 + s23) + (s45 + s67);
    const float y   = 1.0f / (1.0f + expf(-s));

    // emit observed rows (empty range for non-owner lanes)
    for (int j = lo; j < hi; ++j)
        out[(size_t)t * NOBS + j] = y;

    // ---- W = W*(1 + lr*th1*y^2) + (lr*th0*y)*x  (reuses x fragments) ----
    const float tm = fmaf(b_coef * y, y, 1.0f);
    const float u  = a_coef * y;
#pragma unroll
    for (int c = 0; c < CHUNKS; ++c) {
        wfrag[c].x = fmaf(wfrag[c].x, tm, u * xcur[c].x);
        wfrag[c].y = fmaf(wfrag[c].y, tm, u * xcur[c].y);
    }
}

__launch_bounds__(WAVES * 32, 1)
__global__ void hebb_scan_kernel(const float* __restrict__ X,       // [T, NIN]
                                 const float* __restrict__ Winit,   // [NOUT, NIN]
                                 const float* __restrict__ theta,   // [2]
                                 const int*   __restrict__ obs,     // [NOBS] sorted
                                 float*       __restrict__ out)     // [T, NOBS]
{
    __shared__ float part[2][WAVES * ROWS_PER_WG];

    const int tid   = threadIdx.x;
    const int wave  = tid >> 5;       // 0..7
    const int lane  = tid & 31;
    const int kh    = lane >> 4;      // K half-select within a chunk
    const int n     = lane & 15;      // this lane's output row within the block
    const int row   = blockIdx.x * ROWS_PER_WG + n;
    const int kbase = wave * KPW;

    const float lr     = 1.0f / (float)NIN;
    const float a_coef = lr * theta[0];   // scales y x^T
    const float b_coef = lr * theta[1];   // scales y^2 * W

    // ---- load register-resident W fragments (B-operand layout) ----
    // chunk c covers K = kbase + 4c .. +3 ; lane holds K = kbase+4c+2*kh+{0,1}
    v2f wfrag[CHUNKS];
    {
        const float* wp = Winit + (size_t)row * NIN + kbase + 2 * kh;
#pragma unroll
        for (int c = 0; c < CHUNKS; ++c)
            wfrag[c] = *(const v2f*)(wp + 4 * c);
    }

    // ---- observed-index range for this row (wave 0, lanes 0..15 write out) ----
    int lo = 0, hi = 0;
    if (wave == 0 && kh == 0) {
        int l = 0, r = NOBS;                       // lower_bound(row)
        while (l < r) { int m = (l + r) >> 1; if (obs[m] < row) l = m + 1; else r = m; }
        lo = l;
        r = NOBS;                                  // lower_bound(row+1)
        while (l < r) { int m = (l + r) >> 1; if (obs[m] < row + 1) l = m + 1; else r = m; }
        hi = l;
    }

    // ---- preload x_0 fragments ----
    v2f xf0[CHUNKS], xf1[CHUNKS];
    {
        const float* xp = X + kbase + 2 * kh;
#pragma unroll
        for (int c = 0; c < CHUNKS; ++c)
            xf0[c] = *(const v2f*)(xp + 4 * c);
    }

    // ---- sequential scan, unrolled by 2 to ping-pong x register buffers ----
    for (int t = 0; t < T_STEPS; t += 2) {
        scan_step(t,     X, out, part, wfrag, xf0, xf1,
                  wave, kh, n, kbase, lo, hi, a_coef, b_coef);
        scan_step(t + 1, X, out, part, wfrag, xf1, xf0,
                  wave, kh, n, kbase, lo, hi, a_coef, b_coef);
    }
}

extern "C" void kernel_launch(void* const* d_in, const int* in_sizes, int n_in,
                              void* d_out, int out_size, void* d_ws, size_t ws_size,
                              hipStream_t stream) {
    const float* X     = (const float*)d_in[0];   // [2048, 512]
    const float* Winit = (const float*)d_in[1];   // [512, 512]
    const float* theta = (const float*)d_in[2];   // [2]
    const int*   obs   = (const int*)d_in[3];     // [256]
    float*       out   = (float*)d_out;           // [2048, 256]

    hebb_scan_kernel<<<dim3(512 / ROWS_PER_WG), dim3(WAVES * 32), 0, stream>>>(
        X, Winit, theta, obs, out);
}